// DictLearn_61091614818894
// MI455X (gfx1250) — compile-verified
//
#include <hip/hip_runtime.h>
#include <hip/hip_bf16.h>

// DictLearn on MI455X (gfx1250, wave32).
// Phase 1: dual fp32 WMMA GEMM (logits = x@W^T+b, dots = x@dict^T), 128x64 WG tile,
//          double-buffered K pipeline with async global->LDS staging (ASYNCcnt).
// Phase 2: per-row softmax + top-8 distance select + sparse rep/recon + loss partials.
// Phase 3: finalize loss & perplexity.

typedef __attribute__((ext_vector_type(2))) float v2f;
typedef __attribute__((ext_vector_type(8))) float v8f;
typedef __attribute__((__vector_size__(4 * sizeof(int)))) int i4v;

#define NTOK   65536      // B*H*W
#define CDIM   256
#define KDIM   1024
#define TOTAL  16777216   // B*C*H*W

#define WMMA_F32(a, bf, c) \
  __builtin_amdgcn_wmma_f32_16x16x4_f32(false, (a), false, (bf), (short)0, (c), false, false)

#define AS1 __attribute__((address_space(1)))
#define AS3 __attribute__((address_space(3)))

#if __has_builtin(__builtin_amdgcn_global_load_async_to_lds_b128)
#define HAS_ASYNC_LDS 1
#else
#define HAS_ASYNC_LDS 0
#endif

// Copy 16 bytes global -> LDS. Async (no VGPR round-trip, ASYNCcnt) when available.
__device__ __forceinline__ void copy16_g2l(const float* g, float* l) {
#if HAS_ASYNC_LDS
  __builtin_amdgcn_global_load_async_to_lds_b128((AS1 i4v*)g, (AS3 i4v*)l, 0, 0);
#else
  *(float4*)l = *(const float4*)g;
#endif
}

// Wait for all but the most recent 8 async ops (one staged chunk in flight).
__device__ __forceinline__ void async_wait_keep8() {
#if HAS_ASYNC_LDS
#if __has_builtin(__builtin_amdgcn_s_wait_asynccnt)
  __builtin_amdgcn_s_wait_asynccnt(8);
#else
  asm volatile("s_wait_asynccnt 0x8" ::: "memory");
#endif
#endif
}
__device__ __forceinline__ void async_wait_all() {
#if HAS_ASYNC_LDS
#if __has_builtin(__builtin_amdgcn_s_wait_asynccnt)
  __builtin_amdgcn_s_wait_asynccnt(0);
#else
  asm volatile("s_wait_asynccnt 0x0" ::: "memory");
#endif
#endif
}

// ---------------- init small accumulators ----------------
__global__ void dl_init_kernel(float* wsf) {
  int i = blockIdx.x * 256 + threadIdx.x;
  if (i < 1032) wsf[i] = 0.0f;   // [0]=mse, [1]=reg, [8..1031]=counts
}

// ---------------- ||x_n||^2 (coalesced over hw) ----------------
__global__ __launch_bounds__(256) void dl_xsq_kernel(const float* __restrict__ x,
                                                     float* __restrict__ xsq) {
  const int blk  = blockIdx.x;              // 256 blocks
  const int bimg = blk >> 4;                // batch
  const int hw   = (blk & 15) * 256 + threadIdx.x;
  const float* p = x + (size_t)bimg * CDIM * 4096 + hw;
  float acc = 0.0f;
  for (int c = 0; c < CDIM; ++c) { float v = p[(size_t)c * 4096]; acc += v * v; }
  xsq[bimg * 4096 + hw] = acc;
}

// ---------------- ||dict_k||^2 (wave per row) ----------------
__global__ __launch_bounds__(256) void dl_dsq_kernel(const float* __restrict__ dict,
                                                     float* __restrict__ dsq) {
  const int wave = threadIdx.x >> 5, lane = threadIdx.x & 31;
  const int row  = blockIdx.x * 8 + wave;   // 128 blocks * 8 waves = 1024 rows
  float acc = 0.0f;
#pragma unroll
  for (int i = 0; i < 8; ++i) { float v = dict[(size_t)row * CDIM + i * 32 + lane]; acc += v * v; }
#pragma unroll
  for (int s = 16; s > 0; s >>= 1) acc += __shfl_down(acc, s, 32);
  if (lane == 0) dsq[row] = acc;
}

// ---------------- dual fp32 WMMA GEMM ----------------
// Grid: (KDIM/64, NTOK/128). 8 waves: 4(M) x 2(N); each wave: 32x32 per matrix.
// Double-buffered K chunks of 32; 8 async g->LDS ops per thread per chunk.
__global__ __launch_bounds__(256) void dl_gemm_kernel(const float* __restrict__ x,
                                                      const float* __restrict__ W,
                                                      const float* __restrict__ dict,
                                                      const float* __restrict__ bias,
                                                      float* __restrict__ logits,
                                                      float* __restrict__ dots) {
  __shared__ float At[2][32][128];   // k-major A tiles (coalesced, conflict-free frag loads)
  __shared__ float Wt[2][64][36];    // [col][k], rows padded to 144B (16B aligned for b128)
  __shared__ float Dt[2][64][36];

  const int tid  = threadIdx.x;
  const int lane = tid & 31;
  const int wave = tid >> 5;
  const int l16  = lane & 15;
  const int kq   = lane >> 4;

  const int n0   = blockIdx.x * 64;     // output column base (0..KDIM-1)
  const int m0   = blockIdx.y * 128;    // token base
  const int bimg = m0 >> 12;            // 128 | 4096 -> block never crosses batch
  const int hw0  = m0 & 4095;
  const float* xbase = x + (size_t)bimg * CDIM * 4096 + hw0;

  const int wm = (wave >> 1) * 32;      // wave M offset inside tile
  const int wn = (wave & 1) * 32;       // wave N offset inside tile

  // staging assignment (per thread, fixed): A: 4x16B, W: 2x16B, D: 2x16B
  const int akk[4] = { (tid + 0)   >> 5, (tid + 256) >> 5,
                       (tid + 512) >> 5, (tid + 768) >> 5 };
  const int ar4 = tid & 31;
  const int bcol[2] = { (tid + 0) >> 3, (tid + 256) >> 3 };
  const int bk4 = tid & 7;

  v8f accW[2][2], accD[2][2];
  {
    v8f z = {};
#pragma unroll
    for (int i = 0; i < 2; ++i)
#pragma unroll
      for (int j = 0; j < 2; ++j) { accW[i][j] = z; accD[i][j] = z; }
  }

#define STAGE_CHUNK(buf, k0)                                                          \
  do {                                                                                \
    _Pragma("unroll")                                                                 \
    for (int it = 0; it < 4; ++it)                                                    \
      copy16_g2l(xbase + (size_t)((k0) + akk[it]) * 4096 + ar4 * 4,                   \
                 &At[buf][akk[it]][ar4 * 4]);                                         \
    _Pragma("unroll")                                                                 \
    for (int it = 0; it < 2; ++it) {                                                  \
      copy16_g2l(W + (size_t)(n0 + bcol[it]) * CDIM + (k0) + bk4 * 4,                 \
                 &Wt[buf][bcol[it]][bk4 * 4]);                                        \
      copy16_g2l(dict + (size_t)(n0 + bcol[it]) * CDIM + (k0) + bk4 * 4,              \
                 &Dt[buf][bcol[it]][bk4 * 4]);                                        \
    }                                                                                 \
  } while (0)

  STAGE_CHUNK(0, 0);

  for (int c = 0; c < 8; ++c) {
    const int cur = c & 1;
    if (c < 7) {
      STAGE_CHUNK(cur ^ 1, (c + 1) * 32);  // keep next chunk in flight
      async_wait_keep8();                   // drain chunk c, leave c+1 outstanding
    } else {
      async_wait_all();
    }
    __syncthreads();

#pragma unroll
    for (int kk = 0; kk < 8; ++kk) {
      const int k = kk * 4 + 2 * kq;              // lanes 16-31 carry K=2,3 halves
      v2f a0, a1;
      a0.x = At[cur][k][wm + l16];       a0.y = At[cur][k + 1][wm + l16];
      a1.x = At[cur][k][wm + 16 + l16];  a1.y = At[cur][k + 1][wm + 16 + l16];
#pragma unroll
      for (int nt = 0; nt < 2; ++nt) {
        const int col = wn + nt * 16 + l16;
        v2f bw, bd;
        bw.x = Wt[cur][col][k]; bw.y = Wt[cur][col][k + 1];
        bd.x = Dt[cur][col][k]; bd.y = Dt[cur][col][k + 1];
        accW[0][nt] = WMMA_F32(a0, bw, accW[0][nt]);
        accW[1][nt] = WMMA_F32(a1, bw, accW[1][nt]);
        accD[0][nt] = WMMA_F32(a0, bd, accD[0][nt]);
        accD[1][nt] = WMMA_F32(a1, bd, accD[1][nt]);
      }
    }
    __syncthreads();
  }
#undef STAGE_CHUNK

  // Epilogue: bias into logits, write both score matrices.
  // C/D layout: VGPR v -> row = base + v + 8*(lane/16), col = base + lane%16.
#pragma unroll
  for (int mt = 0; mt < 2; ++mt) {
#pragma unroll
    for (int nt = 0; nt < 2; ++nt) {
      const int col   = n0 + wn + nt * 16 + l16;
      const float bv  = bias[col];
      const int rbase = m0 + wm + mt * 16 + 8 * kq;
#pragma unroll
      for (int v = 0; v < 8; ++v) {
        const size_t off = (size_t)(rbase + v) * KDIM + col;
        logits[off] = accW[mt][nt][v] + bv;
        dots[off]   = accD[mt][nt][v];
      }
    }
  }
}

// ---------------- per-row: softmax, top-8, rep, recon, loss partials ----------------
__global__ __launch_bounds__(256) void dl_row_kernel(const float* __restrict__ xlin,
                                                     const float* __restrict__ dict,
                                                     const float* __restrict__ xsq,
                                                     const float* __restrict__ dsq,
                                                     const float* __restrict__ dots,
                                                     float* __restrict__ rep,   // holds logits on entry
                                                     float* __restrict__ recon,
                                                     float* msum, float* rsum,
                                                     unsigned* counts) {
  __shared__ float lg[KDIM];
  __shared__ float dst[KDIM];
  __shared__ float rv[256];
  __shared__ int   ri[256];
  __shared__ int   selIdx[8];
  __shared__ float selVal[8];
  __shared__ float srowmax, srowsum;

  const int n   = blockIdx.x;
  const int tid = threadIdx.x;
  const float xs = xsq[n];

  float lmax = -3.4e38f;
#pragma unroll
  for (int j = 0; j < 4; ++j) {
    const int c = tid + j * 256;
    const float L = rep[(size_t)n * KDIM + c];
    lg[c]  = L;
    dst[c] = xs + dsq[c] - 2.0f * dots[(size_t)n * KDIM + c];
    lmax = fmaxf(lmax, L);
  }
  rv[tid] = lmax; __syncthreads();
  for (int s = 128; s > 0; s >>= 1) { if (tid < s) rv[tid] = fmaxf(rv[tid], rv[tid + s]); __syncthreads(); }
  if (tid == 0) srowmax = rv[0];
  __syncthreads();
  const float mx = srowmax;

  float lsum = 0.0f;
#pragma unroll
  for (int j = 0; j < 4; ++j) lsum += __expf(lg[tid + j * 256] - mx);
  rv[tid] = lsum; __syncthreads();
  for (int s = 128; s > 0; s >>= 1) { if (tid < s) rv[tid] += rv[tid + s]; __syncthreads(); }
  if (tid == 0) srowsum = rv[0];
  __syncthreads();
  const float inv = 1.0f / srowsum;

  // 8 rounds of argmin over the 1024 distances
  for (int t = 0; t < 8; ++t) {
    float best = 3.4e38f; int bi = 0;
#pragma unroll
    for (int j = 0; j < 4; ++j) {
      const int c = tid + j * 256;
      const float d = dst[c];
      if (d < best) { best = d; bi = c; }
    }
    rv[tid] = best; ri[tid] = bi; __syncthreads();
    for (int s = 128; s > 0; s >>= 1) {
      if (tid < s && rv[tid + s] < rv[tid]) { rv[tid] = rv[tid + s]; ri[tid] = ri[tid + s]; }
      __syncthreads();
    }
    if (tid == 0) {
      const int w = ri[0];
      selIdx[t] = w;
      selVal[t] = __expf(lg[w] - mx) * inv;
      dst[w] = 3.4e38f;
    }
    __syncthreads();
  }

  // sparse rep row (overwrites the logits that lived here)
#pragma unroll
  for (int j = 0; j < 4; ++j) {
    const int c = tid + j * 256;
    float v = 0.0f;
#pragma unroll
    for (int t = 0; t < 8; ++t) if (c == selIdx[t]) v = selVal[t];
    rep[(size_t)n * KDIM + c] = v;
  }

  // recon row (8 atoms) + mse partial (recon_out is the flat (N,C) buffer per .view)
  float part;
  {
    const int c = tid;  // one channel per thread
    float r = 0.0f;
#pragma unroll
    for (int t = 0; t < 8; ++t) r += selVal[t] * dict[(size_t)selIdx[t] * CDIM + c];
    const size_t off = (size_t)n * CDIM + c;
    recon[off] = r;
    const float d = xlin[off] - r;
    part = d * d;
  }
  rv[tid] = part; __syncthreads();
  for (int s = 128; s > 0; s >>= 1) { if (tid < s) rv[tid] += rv[tid + s]; __syncthreads(); }
  if (tid == 0) {
    atomicAdd(msum, rv[0]);
    float rs = 0.0f;
#pragma unroll
    for (int t = 0; t < 8; ++t) rs += selVal[t];
    atomicAdd(rsum, rs);
  }
  if (tid < 8) atomicAdd(&counts[selIdx[tid]], 1u);
}

// ---------------- finalize: loss & perplexity ----------------
__global__ __launch_bounds__(256) void dl_final_kernel(const unsigned* __restrict__ counts,
                                                       const float* msum, const float* rsum,
                                                       float* out_loss, float* out_perp) {
  __shared__ float red[256];
  __shared__ unsigned redu[256];
  const int tid = threadIdx.x;
  unsigned csum = 0;
#pragma unroll
  for (int j = 0; j < 4; ++j) csum += counts[tid + j * 256];
  redu[tid] = csum; __syncthreads();
  for (int s = 128; s > 0; s >>= 1) { if (tid < s) redu[tid] += redu[tid + s]; __syncthreads(); }
  const float total = (float)redu[0];
  float ent = 0.0f;
#pragma unroll
  for (int j = 0; j < 4; ++j) {
    const float p = (float)counts[tid + j * 256] / total;
    ent += p * __logf(p + 1e-10f);
  }
  red[tid] = ent; __syncthreads();
  for (int s = 128; s > 0; s >>= 1) { if (tid < s) red[tid] += red[tid + s]; __syncthreads(); }
  if (tid == 0) {
    *out_perp = __expf(-red[0]);
    *out_loss = 2.0f * (*msum) / (float)TOTAL + (*rsum);
  }
}

extern "C" void kernel_launch(void* const* d_in, const int* in_sizes, int n_in,
                              void* d_out, int out_size, void* d_ws, size_t ws_size,
                              hipStream_t stream) {
  const float* x    = (const float*)d_in[0];
  const float* dict = (const float*)d_in[1];
  const float* W    = (const float*)d_in[2];
  const float* b    = (const float*)d_in[3];

  float* out      = (float*)d_out;
  float* out_loss = out;                       // [0]
  float* recon    = out + 1;                   // [1 .. TOTAL]
  float* out_perp = out + 1 + TOTAL;           // [TOTAL+1]
  float* rep      = out + 2 + TOTAL;           // [TOTAL+2 ..), N*K floats

  // workspace layout (floats): 0 mse, 1 reg, 8..1031 counts,
  // 1032.. xsq(65536), 66568.. dsq(1024), 67592.. dots(N*K) -> ~269 MB total
  float*    wsf    = (float*)d_ws;
  float*    msum   = wsf + 0;
  float*    rsum   = wsf + 1;
  unsigned* counts = (unsigned*)(wsf + 8);
  float*    xsq    = wsf + 1032;
  float*    dsq    = wsf + 1032 + NTOK;
  float*    dots   = wsf + 1032 + NTOK + KDIM;

  dl_init_kernel<<<5, 256, 0, stream>>>(wsf);
  dl_xsq_kernel<<<256, 256, 0, stream>>>(x, xsq);
  dl_dsq_kernel<<<128, 256, 0, stream>>>(dict, dsq);
  dl_gemm_kernel<<<dim3(KDIM / 64, NTOK / 128), 256, 0, stream>>>(x, W, dict, b, rep, dots);
  dl_row_kernel<<<NTOK, 256, 0, stream>>>(x, dict, xsq, dsq, dots, rep, recon,
                                          msum, rsum, counts);
  dl_final_kernel<<<1, 256, 0, stream>>>(counts, msum, rsum, out_loss, out_perp);
}